// EMA_36902359007714
// MI455X (gfx1250) — compile-verified
//
#include <hip/hip_runtime.h>

typedef __attribute__((ext_vector_type(2))) float v2f;
typedef __attribute__((ext_vector_type(4))) float v4f;
typedef __attribute__((ext_vector_type(8))) float v8f;

#define LSEQ 4096
#define BD   2048          // B*D = 16*128
#define EMAS 8
#define CH   16            // chunk length (time) == WMMA M
#define NCH  (LSEQ / CH)   // 256 chunks
#define NCT  (BD / 16)     // 128 column tiles
#define TPW  8             // time-tiles per wave in kernel C

// ---------------------------------------------------------------------------
// Kernel A: per-chunk partial end state with zero carry-in.
// S[c][e][j] = sum_{k=0..15} d_e (1-d_e)^{15-k} x[16c+k][j]
// One thread per (chunk, column). Coalesced loads/stores.
// ---------------------------------------------------------------------------
__global__ void __launch_bounds__(256)
ema_chunk_state(const float* __restrict__ x,
                const float* __restrict__ log_decay,
                float* __restrict__ cb) {
    const int t = blockIdx.x * 256 + threadIdx.x;   // 0 .. NCH*BD-1
    const int j = t & (BD - 1);
    const int c = t >> 11;                          // t / 2048

    float d[EMAS], omd[EMAS], s[EMAS];
#pragma unroll
    for (int e = 0; e < EMAS; ++e) {
        const float ld = log_decay[e];
        d[e]   = 1.0f / (1.0f + __expf(-ld));
        omd[e] = 1.0f - d[e];
        s[e]   = 0.0f;
    }
    const float* xp = x + (size_t)c * CH * BD + j;
#pragma unroll
    for (int k = 0; k < CH; ++k) {
        const float xv = xp[(size_t)k * BD];
#pragma unroll
        for (int e = 0; e < EMAS; ++e)
            s[e] = fmaf(omd[e], s[e], d[e] * xv);
    }
    float* sp = cb + (size_t)c * EMAS * BD + j;
#pragma unroll
    for (int e = 0; e < EMAS; ++e) sp[(size_t)e * BD] = s[e];
}

// ---------------------------------------------------------------------------
// Kernel B: sequential prefix over the 256 chunk carries (in-place in cb).
// carry_0 = x[0][j];  carry_{c+1} = (1-d)^16 * carry_c + S_c
// After this kernel, cb[c][e][j] holds carry_c (state entering chunk c).
// ---------------------------------------------------------------------------
__global__ void __launch_bounds__(256)
ema_carry_scan(const float* __restrict__ x,
               const float* __restrict__ log_decay,
               float* __restrict__ cb) {
    const int t = blockIdx.x * 256 + threadIdx.x;   // 0 .. EMAS*BD-1
    const int j = t & (BD - 1);
    const int e = t >> 11;

    const float ld  = log_decay[e];
    const float d   = 1.0f / (1.0f + __expf(-ld));
    const float omd = 1.0f - d;
    const float p2 = omd * omd, p4 = p2 * p2, p8 = p4 * p4, p16 = p8 * p8;

    float carry = x[j];                              // init state = x[0] row
    for (int c = 0; c < NCH; ++c) {
        float* slot = cb + ((size_t)c * EMAS + e) * BD + j;
        const float hold = *slot;                    // S_c
        *slot = carry;                               // publish carry_c
        carry = fmaf(p16, carry, hold);              // carry_{c+1}
    }
}

// ---------------------------------------------------------------------------
// Kernel C: main pass. Each 16-timestep x 16-column tile:
//   out[i][j] = sum_{k<=i} d(1-d)^{i-k} x[k][j]  +  (1-d)^{i+1} carry[j]
// computed as 5x V_WMMA_F32_16X16X4_F32 (K = 16 data cols + 1 carry col).
// All 8 EMA channels accumulated per tile -> b128-coalesced stores on the
// innermost EMAS axis. One wave (32 threads) per tile strip.
// ---------------------------------------------------------------------------
__global__ void __launch_bounds__(32)
ema_wmma_main(const float* __restrict__ x,
              const float* __restrict__ log_decay,
              const float* __restrict__ cb,
              float* __restrict__ out) {
    const int lane = threadIdx.x;
    const int i    = lane & 15;      // A row (M) / B,D column (N)
    const int hi   = lane >> 4;      // K half selector for A/B operands
    const int c0   = blockIdx.x * 16;

    // --- Build A operands (per-ema coefficient matrix), hoisted per wave ---
    // A layout (16x4 f32): lane%16 = M; VGPR r holds K = kk*4 + 2*hi + r.
    v2f A[EMAS][5];
    const float fi = (float)i;
#pragma unroll
    for (int e = 0; e < EMAS; ++e) {
        const float ld = log_decay[e];
        const float d  = 1.0f / (1.0f + __expf(-ld));
        const float lg = __builtin_amdgcn_logf(1.0f - d);    // log2(1-d)
#pragma unroll
        for (int kk = 0; kk < 4; ++kk) {
#pragma unroll
            for (int r = 0; r < 2; ++r) {
                const int k = kk * 4 + 2 * hi + r;
                // (1-d)^(i-k) via exp2(log2(1-d) * (i-k))
                const float pw = __builtin_amdgcn_exp2f(lg * (fi - (float)k));
                A[e][kk][r] = (k <= i) ? d * pw : 0.0f;
            }
        }
        // carry column: global K=16 -> local k=0 (hi==0, r==0); rest zero
        A[e][4][0] = (hi == 0) ? __builtin_amdgcn_exp2f(lg * (fi + 1.0f)) : 0.0f;
        A[e][4][1] = 0.0f;
    }

#pragma unroll 1
    for (int tt = 0; tt < TPW; ++tt) {
        const int c  = blockIdx.y * TPW + tt;        // chunk index
        const int t0 = c * CH;

        // --- B operand: x tile, B layout (4x16): lane%16 = N, K split by hi
        v2f Bx[4];
        const float* xp = x + (size_t)t0 * BD + c0 + i;
#pragma unroll
        for (int kk = 0; kk < 4; ++kk) {
            const int kb = kk * 4 + 2 * hi;
            Bx[kk][0] = xp[(size_t)kb * BD];
            Bx[kk][1] = xp[(size_t)(kb + 1) * BD];
        }

        v8f D[EMAS];
#pragma unroll
        for (int e = 0; e < EMAS; ++e) {
            v2f Bc;                                   // carry row (K=16)
            Bc[0] = cb[((size_t)c * EMAS + e) * BD + c0 + i];
            Bc[1] = 0.0f;
            v8f acc = {};
#pragma unroll
            for (int kk = 0; kk < 4; ++kk)
                acc = __builtin_amdgcn_wmma_f32_16x16x4_f32(
                    false, A[e][kk], false, Bx[kk], (short)0, acc, false, false);
            acc = __builtin_amdgcn_wmma_f32_16x16x4_f32(
                false, A[e][4], false, Bc, (short)0, acc, false, false);
            D[e] = acc;
        }

        // --- Store: D layout: lane%16 = N col, row M = r + 8*hi.
        // out[l][col][e] with e innermost -> two b128 per row, fully coalesced.
        float* op = out + (((size_t)(t0 + 8 * hi)) * BD + c0 + i) * EMAS;
#pragma unroll
        for (int r = 0; r < 8; ++r) {
            float* rowp = op + (size_t)r * BD * EMAS;
            v4f lo = { D[0][r], D[1][r], D[2][r], D[3][r] };
            v4f h4 = { D[4][r], D[5][r], D[6][r], D[7][r] };
            *(v4f*)(rowp)     = lo;
            *(v4f*)(rowp + 4) = h4;
        }
    }
}

// ---------------------------------------------------------------------------
extern "C" void kernel_launch(void* const* d_in, const int* in_sizes, int n_in,
                              void* d_out, int out_size, void* d_ws, size_t ws_size,
                              hipStream_t stream) {
    (void)in_sizes; (void)n_in; (void)out_size; (void)ws_size;
    const float* x  = (const float*)d_in[0];
    const float* ld = (const float*)d_in[1];
    float* out = (float*)d_out;
    float* cb  = (float*)d_ws;   // NCH*EMAS*BD floats = 16 MB scratch

    ema_chunk_state<<<dim3((NCH * BD) / 256), dim3(256), 0, stream>>>(x, ld, cb);
    ema_carry_scan <<<dim3((EMAS * BD) / 256), dim3(256), 0, stream>>>(x, ld, cb);
    ema_wmma_main  <<<dim3(NCT, NCH / TPW), dim3(32), 0, stream>>>(x, ld, cb, out);
}